// SPCModel_87308095193448
// MI455X (gfx1250) — compile-verified
//
#include <hip/hip_runtime.h>
#include <hip/hip_bf16.h>

// FWHT(65536) = H16 x H16 x H16 x H16 applied as four radix-16 stages.
// One workgroup per (batch,channel) row; whole row resident in 288KB LDS.
// Each radix-16 stage: 256 tiles of 16x16, each tile = H16 * M via four
// chained V_WMMA_F32_16X16X4_F32 (fp32-exact, since H16 entries are +-1).

typedef __attribute__((ext_vector_type(2))) float v2f;
typedef __attribute__((ext_vector_type(8))) float v8f;

#define N_TOTAL   65536          // 16^4
#define LDS_FLOATS (N_TOTAL + (N_TOTAL / 16) * 2)   // pad 2 floats per 16 -> 73728 floats = 288KB
#define LDS_BYTES  (LDS_FLOATS * 4)

__device__ __forceinline__ int pidx(int n) { return n + ((n >> 4) << 1); }

__global__ __launch_bounds__(256) void fwht_mask_kernel(
    const float* __restrict__ x,
    const float* __restrict__ mask,
    float* __restrict__ out)
{
    extern __shared__ float lds[];

    const int row  = blockIdx.x;
    const float* __restrict__ xin  = x   + (size_t)row * N_TOTAL;
    float*       __restrict__ xout = out + (size_t)row * N_TOTAL;

    const int tid  = threadIdx.x;     // 0..255  (8 wave32 waves)
    const int lane = tid & 31;
    const int wave = tid >> 5;

    // ---------------- global -> padded LDS (one pass) ----------------
    // float4 global loads; each float4 stays inside one 16-float block,
    // so the padded destination is contiguous (two 8B-aligned v2f stores).
    for (int i = tid; i < N_TOTAL / 4; i += 256) {
        const float4 v = ((const float4*)xin)[i];
        const int p = pidx(i * 4);
        *(v2f*)&lds[p]     = v2f{v.x, v.y};
        *(v2f*)&lds[p + 2] = v2f{v.z, v.w};
    }
    __syncthreads();

    // ---------------- per-lane H16 operand (A matrix) ----------------
    // A is 16x4 f32: lanes 0-15 hold K={0,1}, lanes 16-31 hold K={2,3}.
    // Chunk q supplies columns H16[:, 4q..4q+3].  H16[i][j]=(-1)^popc(i&j).
    const int mrow  = lane & 15;
    const int khalf = (lane >> 4) << 1;
    v2f A[4];
#pragma unroll
    for (int q = 0; q < 4; ++q) {
        const int k0 = 4 * q + khalf;
        A[q].x = (__popc(mrow & k0)       & 1) ? -1.0f : 1.0f;
        A[q].y = (__popc(mrow & (k0 + 1)) & 1) ? -1.0f : 1.0f;
    }

    const int cc    = lane & 15;        // B/D column index within tile
    const int jbase = (lane >> 4) << 3; // D rows: 0..7 or 8..15

    // ---------------- four radix-16 stages over LDS ----------------
#pragma unroll
    for (int s = 0; s < 4; ++s) {
        const int sh = 4 * s;           // stride S = 1 << sh
        for (int t = 0; t < 32; ++t) {
            const int T = (wave << 5) + t;     // tile 0..255 (wave-uniform)

            // element index n(j, cc) = base + j*rstride + cc*cstride
            int base, rstride, cstride;
            if (s == 0) {
                base = T << 8; rstride = 1; cstride = 16;
            } else {
                const int c0 = T << 4;
                const int hi = c0 >> sh;            // S >= 16 here
                const int lo = c0 & ((1 << sh) - 1);
                base = (hi << (sh + 4)) + lo;
                rstride = 1 << sh; cstride = 1;
            }
            const int coff = cc * cstride;

            // B is 4x16 per chunk: lanes 0-15 hold K={4q,4q+1},
            // lanes 16-31 hold K={4q+2,4q+3}  (mirrors the A split).
            v8f C = {0.f, 0.f, 0.f, 0.f, 0.f, 0.f, 0.f, 0.f};
#pragma unroll
            for (int q = 0; q < 4; ++q) {
                const int k0 = 4 * q + khalf;
                v2f Bq;
                Bq.x = lds[pidx(base + k0 * rstride + coff)];
                Bq.y = lds[pidx(base + (k0 + 1) * rstride + coff)];
                C = __builtin_amdgcn_wmma_f32_16x16x4_f32(
                        /*neg_a=*/false, A[q],
                        /*neg_b=*/false, Bq,
                        /*c_mod=*/(short)0, C,
                        /*reuse_a=*/false, /*reuse_b=*/false);
            }

            // D (16x16 f32, 8 VGPRs): reg r holds row jbase+r at column cc.
            // Write set == read set of this tile -> safe in-place per wave.
#pragma unroll
            for (int r = 0; r < 8; ++r) {
                lds[pidx(base + (jbase + r) * rstride + coff)] = C[r];
            }
        }
        __syncthreads();
    }

    // ---------------- padded LDS -> global, fused mask * 1/sqrt(N) ----------------
    for (int i = tid; i < N_TOTAL / 4; i += 256) {
        const int p = pidx(i * 4);
        const float4 m = ((const float4*)mask)[i];
        float4 o;
        o.x = lds[p]     * m.x * (1.0f / 256.0f);
        o.y = lds[p + 1] * m.y * (1.0f / 256.0f);
        o.z = lds[p + 2] * m.z * (1.0f / 256.0f);
        o.w = lds[p + 3] * m.w * (1.0f / 256.0f);
        ((float4*)xout)[i] = o;
    }
}

extern "C" void kernel_launch(void* const* d_in, const int* in_sizes, int n_in,
                              void* d_out, int out_size, void* d_ws, size_t ws_size,
                              hipStream_t stream) {
    const float* x    = (const float*)d_in[0];   // (128, 3, 256, 256) fp32
    const float* mask = (const float*)d_in[1];   // (65536,) fp32
    float* out        = (float*)d_out;           // (128, 3, 65536) fp32

    const int rows = in_sizes[0] / N_TOTAL;      // 128*3 = 384
    fwht_mask_kernel<<<rows, 256, LDS_BYTES, stream>>>(x, mask, out);
}